// BPSDensenet_46282567582129
// MI455X (gfx1250) — compile-verified
//
#include <hip/hip_runtime.h>

typedef __attribute__((ext_vector_type(2))) float v2f;
typedef __attribute__((ext_vector_type(8))) float v8f;

#define BB 32
#define NN 2048
#define PP 1024
#define HH 256
#define EE 512
#define BN_EPS 1e-5f

// ---------------------------------------------------------------------------
// Kernel 1: BPS encoding via V_WMMA_F32_16X16X4_F32.
//   A (16x4) = (x, y, z, |p|^2) per point row
//   B (4x16) = (-2bx, -2by, -2bz, 1) per basis column
//   D = |p|^2 - 2 p.b  ;  dist = D + |b|^2 (per-column constant, added after min)
// Grid: (P/16, B), Block: 256 (8 waves); each wave owns the 16-col tile,
// waves split the N=2048 points, LDS min-tree combines.
// ---------------------------------------------------------------------------
__global__ __launch_bounds__(256) void bps_min_kernel(
    const float* __restrict__ x,      // [B, 3, N]
    const float* __restrict__ basis,  // [P, 3]
    float* __restrict__ feat)         // [B, P]
{
  const int tileP = blockIdx.x * 16;
  const int b     = blockIdx.y;
  const int lane  = threadIdx.x & 31;
  const int wave  = threadIdx.x >> 5;
  const int col   = lane & 15;
  const bool hi   = lane >= 16;

  // Basis tile -> B-matrix registers (constant for whole kernel)
  const int   n   = tileP + col;
  const float bx  = basis[n * 3 + 0];
  const float by  = basis[n * 3 + 1];
  const float bz  = basis[n * 3 + 2];
  const float bn2 = bx * bx + by * by + bz * bz;
  v2f Bm;
  Bm.x = hi ? (-2.0f * bz) : (-2.0f * bx);  // K=2 : K=0
  Bm.y = hi ? 1.0f         : (-2.0f * by);  // K=3 : K=1

  // 4 running minima (each folds 2 of the 8 D rows via a min3 tree)
  float minv[4];
#pragma unroll
  for (int r = 0; r < 4; ++r) minv[r] = 3.4e38f;

  const float* xb = x + (size_t)b * 3 * NN;
#pragma unroll 2
  for (int i0 = wave * 16; i0 < NN; i0 += 16 * 8) {
    const int   m   = i0 + col;
    const float px  = xb[m];
    const float py  = xb[NN + m];
    const float pz  = xb[2 * NN + m];
    const float pn2 = px * px + py * py + pz * pz;
    v2f Am;
    Am.x = hi ? pz  : px;   // K=2 : K=0
    Am.y = hi ? pn2 : py;   // K=3 : K=1
    v8f c = {0.f, 0.f, 0.f, 0.f, 0.f, 0.f, 0.f, 0.f};
    c = __builtin_amdgcn_wmma_f32_16x16x4_f32(false, Am, false, Bm,
                                              (short)0, c, false, false);
#pragma unroll
    for (int r = 0; r < 4; ++r)                       // v_min3 tree
      minv[r] = fminf(fminf(c[2 * r], c[2 * r + 1]), minv[r]);
  }
  float mv = fminf(fminf(minv[0], minv[1]), fminf(minv[2], minv[3]));
  mv = fminf(mv, __shfl_xor(mv, 16));  // combine M and M+8 lane halves
  mv = fmaxf(mv + bn2, 0.0f);          // clamp(d, 0)

  __shared__ float red[8][16];
  if (lane < 16) red[wave][col] = mv;
  __syncthreads();
  if (threadIdx.x < 16) {
    float v = red[0][threadIdx.x];
#pragma unroll
    for (int w = 1; w < 8; ++w) v = fminf(v, red[w][threadIdx.x]);
    feat[(size_t)b * PP + tileP + threadIdx.x] = sqrtf(v);
  }
}

// ---------------------------------------------------------------------------
// Kernel 2: BatchNorm over the batch dim for feat -> x0. One thread per column.
// ---------------------------------------------------------------------------
__global__ __launch_bounds__(256) void bn0_kernel(
    const float* __restrict__ feat, const float* __restrict__ g,
    const float* __restrict__ bshift, float* __restrict__ x0)
{
  const int j = blockIdx.x * 256 + threadIdx.x;
  float s = 0.f, ss = 0.f;
#pragma unroll 4
  for (int i = 0; i < BB; ++i) {
    const float v = feat[(size_t)i * PP + j];
    s += v; ss += v * v;
  }
  const float mean = s * (1.0f / BB);
  const float var  = ss * (1.0f / BB) - mean * mean;
  const float sc   = g[j] * rsqrtf(var + BN_EPS);
  const float bt   = bshift[j];
#pragma unroll 4
  for (int i = 0; i < BB; ++i)
    x0[(size_t)i * PP + j] = (feat[(size_t)i * PP + j] - mean) * sc + bt;
}

// ---------------------------------------------------------------------------
// One concat segment of the GEMM: acc += in[rows, Ks] @ Wseg.T tile.
// Tight body: 2x float2 load + 1 wmma, no per-iteration segment select.
// ---------------------------------------------------------------------------
__device__ __forceinline__ void gemm_segment(v8f& acc, const float* __restrict__ src,
                                             int Ks, const float* __restrict__ Wseg,
                                             int row, int koff) {
  const float* arow = src + (size_t)row * Ks + koff;
  const float* wrow = Wseg + koff;
#pragma unroll 4
  for (int kk = 0; kk < Ks; kk += 4) {
    const float2 av = *reinterpret_cast<const float2*>(arow + kk);
    const float2 wv = *reinterpret_cast<const float2*>(wrow + kk);
    v2f Am; Am.x = av.x; Am.y = av.y;
    v2f Bm; Bm.x = wv.x; Bm.y = wv.y;
    acc = __builtin_amdgcn_wmma_f32_16x16x4_f32(false, Am, false, Bm,
                                                (short)0, acc, false, false);
  }
}

// ---------------------------------------------------------------------------
// Kernel 3: fused  out = BN(relu(concat(in0,in1,in2) @ W.T + bias))
// Grid: Hout/16 blocks of 64 threads (2 waves = 2 M-tiles of the 32 rows).
// Each workgroup owns all 32 batch rows of its 16 columns, so BN batch stats
// reduce fully inside the workgroup (registers -> shfl_xor(16) -> LDS).
// ---------------------------------------------------------------------------
__global__ __launch_bounds__(64) void dense_bn_relu_kernel(
    const float* __restrict__ in0, int K0,
    const float* __restrict__ in1, int K1,
    const float* __restrict__ in2, int K2,
    const float* __restrict__ W,     // [Hout, Ktot] row-major
    const float* __restrict__ bias,
    const float* __restrict__ g, const float* __restrict__ beta,
    float* __restrict__ out, int Hout)
{
  const int  c    = blockIdx.x * 16;
  const int  lane = threadIdx.x & 31;
  const int  wave = threadIdx.x >> 5;
  const int  col  = lane & 15;
  const bool hi   = lane >= 16;
  const int  n    = c + col;
  const int  row  = wave * 16 + col;
  const int  Ktot = K0 + K1 + K2;
  const int  koff = hi ? 2 : 0;  // lanes 16-31 hold K+2, K+3

  v8f acc = {0.f, 0.f, 0.f, 0.f, 0.f, 0.f, 0.f, 0.f};
  const float* Wn = W + (size_t)n * Ktot;  // W.T column n
  gemm_segment(acc, in0, K0, Wn, row, koff);
  if (K1 > 0) gemm_segment(acc, in1, K1, Wn + K0, row, koff);
  if (K2 > 0) gemm_segment(acc, in2, K2, Wn + K0 + K1, row, koff);

  // bias + relu, partial batch stats (8 rows per lane)
  const float bi = bias[n];
  float vals[8];
  float s = 0.f, ss = 0.f;
#pragma unroll
  for (int r = 0; r < 8; ++r) {
    const float v = fmaxf(acc[r] + bi, 0.f);
    vals[r] = v; s += v; ss += v * v;
  }
  s  += __shfl_xor(s, 16);   // combine M and M+8 halves (same column)
  ss += __shfl_xor(ss, 16);

  __shared__ float sred[2][16][2];
  if (lane < 16) { sred[wave][col][0] = s; sred[wave][col][1] = ss; }
  __syncthreads();
  const float ts   = sred[0][col][0] + sred[1][col][0];
  const float tss  = sred[0][col][1] + sred[1][col][1];
  const float mean = ts * (1.0f / BB);
  const float var  = tss * (1.0f / BB) - mean * mean;
  const float sc   = g[n] * rsqrtf(var + BN_EPS);
  const float bt   = beta[n];
#pragma unroll
  for (int r = 0; r < 8; ++r) {
    const int rr = wave * 16 + r + (hi ? 8 : 0);
    out[(size_t)rr * Hout + n] = (vals[r] - mean) * sc + bt;
  }
}

// ---------------------------------------------------------------------------
extern "C" void kernel_launch(void* const* d_in, const int* in_sizes, int n_in,
                              void* d_out, int out_size, void* d_ws, size_t ws_size,
                              hipStream_t stream) {
  const float* x     = (const float*)d_in[0];
  const float* basis = (const float*)d_in[1];
  const float* bn0_g = (const float*)d_in[2];
  const float* bn0_b = (const float*)d_in[3];
  const float* W0 = (const float*)d_in[4];
  const float* b0 = (const float*)d_in[5];
  const float* g0 = (const float*)d_in[6];
  const float* be0 = (const float*)d_in[7];
  const float* W1 = (const float*)d_in[8];
  const float* b1 = (const float*)d_in[9];
  const float* g1 = (const float*)d_in[10];
  const float* be1 = (const float*)d_in[11];
  const float* W2 = (const float*)d_in[12];
  const float* b2 = (const float*)d_in[13];
  const float* g2 = (const float*)d_in[14];
  const float* be2 = (const float*)d_in[15];
  const float* W3 = (const float*)d_in[16];
  const float* b3 = (const float*)d_in[17];
  const float* g3 = (const float*)d_in[18];
  const float* be3 = (const float*)d_in[19];
  const float* Wf = (const float*)d_in[20];
  const float* bf = (const float*)d_in[21];
  const float* gf = (const float*)d_in[22];
  const float* bef = (const float*)d_in[23];

  float* ws   = (float*)d_ws;
  float* feat = ws;           // 32*1024
  float* x0   = ws + 32768;   // 32*1024
  float* h    = ws + 65536;   // 32*256
  float* a1   = ws + 73728;   // 32*256
  float* a2   = ws + 81920;   // 32*256

  // 1) BPS encoding (WMMA distance tiles + min reduction)
  bps_min_kernel<<<dim3(PP / 16, BB), 256, 0, stream>>>(x, basis, feat);
  // 2) x0 = BN(feat)
  bn0_kernel<<<PP / 256, 256, 0, stream>>>(feat, bn0_g, bn0_b, x0);
  // 3) stage 0
  dense_bn_relu_kernel<<<HH / 16, 64, 0, stream>>>(x0, PP, nullptr, 0, nullptr, 0,
                                                   W0, b0, g0, be0, h, HH);
  dense_bn_relu_kernel<<<HH / 16, 64, 0, stream>>>(h, HH, nullptr, 0, nullptr, 0,
                                                   W1, b1, g1, be1, a1, HH);
  // 4) stage 1 (dense concat [x0, a1])
  dense_bn_relu_kernel<<<HH / 16, 64, 0, stream>>>(x0, PP, a1, HH, nullptr, 0,
                                                   W2, b2, g2, be2, h, HH);
  dense_bn_relu_kernel<<<HH / 16, 64, 0, stream>>>(h, HH, nullptr, 0, nullptr, 0,
                                                   W3, b3, g3, be3, a2, HH);
  // 5) head (concat [x0, a1, a2])
  dense_bn_relu_kernel<<<EE / 16, 64, 0, stream>>>(x0, PP, a1, HH, a2, HH,
                                                   Wf, bf, gf, bef,
                                                   (float*)d_out, EE);
}